// PerNodeSimpleAnalyticBayesVelocityEnergy_29162827940079
// MI455X (gfx1250) — compile-verified
//
#include <hip/hip_runtime.h>
#include <hip/hip_bf16.h>
#include <math.h>

// Problem constants (from reference): S=128 samples, N=2048 rows, D=64 nodes.
#define S_DIM 128
#define N_ROWS 2048
#define D_DIM 64
#define BETA 0.01f
#define PRIOR_LAMBDA 0.1f

typedef __attribute__((ext_vector_type(2))) float v2f;
typedef __attribute__((ext_vector_type(8))) float v8f;
typedef __attribute__((ext_vector_type(4))) unsigned int v4u;
typedef __attribute__((ext_vector_type(4))) int v4i;
typedef __attribute__((ext_vector_type(8))) int v8i;

// Feature detection for CDNA5 data-mover paths.
#if defined(__has_builtin)
#  if __has_builtin(__builtin_amdgcn_tensor_load_to_lds) && \
      __has_builtin(__builtin_amdgcn_s_wait_tensorcnt)
#    define HAVE_TDM 1
#  endif
#endif

// Workspace layout (floats):
//   [0      .. 4095 ]  XtX   (64x64)   = x^T x
//   [4096   .. 8191 ]  Xtdx  (64x64)   = x^T dx
//   [8192]             dx2             = sum(dx^2)
//   [12288  .. +S*4096] W              = w[s][p][i] solutions
// Total ~2.15 MB.
#define WS_XTX  0
#define WS_XTDX 4096
#define WS_DX2  8192
#define WS_W    12288

// ---------------------------------------------------------------------------
// Kernel 1: XtX, Xtdx via WMMA f32 16x16x4 (K=2048), plus sum(dx^2).
// Blocks 0..15  -> XtX tile (ti,tj); blocks 16..31 -> Xtdx tile; block 32 -> dx2.
// One wave (32 threads) per block.
// ---------------------------------------------------------------------------
__global__ void k_precompute(const float* __restrict__ x,
                             const float* __restrict__ dx,
                             float* __restrict__ ws) {
    const int b    = blockIdx.x;
    const int lane = threadIdx.x;            // 0..31 (wave32)
    if (b < 32) {
        const bool isXtX = (b < 16);
        const int  t  = isXtX ? b : (b - 16);
        const int  ti = t >> 2;
        const int  tj = t & 3;
        const float* Bsrc = isXtX ? x : dx;
        const int half = lane >> 4;          // 0 or 1
        const int lm   = lane & 15;
        v8f c = {};
        for (int kb = 0; kb < N_ROWS; kb += 4) {
            const int r0 = kb + 2 * half;
            v2f a, bb;
            a.x  = x[(size_t)r0 * D_DIM + ti * 16 + lm];
            a.y  = x[(size_t)(r0 + 1) * D_DIM + ti * 16 + lm];
            bb.x = Bsrc[(size_t)r0 * D_DIM + tj * 16 + lm];
            bb.y = Bsrc[(size_t)(r0 + 1) * D_DIM + tj * 16 + lm];
            c = __builtin_amdgcn_wmma_f32_16x16x4_f32(
                    false, a, false, bb, (short)0, c, false, false);
        }
        float* dst = ws + (isXtX ? WS_XTX : WS_XTDX);
        #pragma unroll
        for (int r = 0; r < 8; ++r) {
            const int m = ti * 16 + r + 8 * half;
            const int n = tj * 16 + lm;
            dst[m * D_DIM + n] = c[r];
        }
    } else {
        float acc = 0.0f;
        for (int i = lane; i < N_ROWS * D_DIM; i += 32) {
            const float v = dx[i];
            acc += v * v;
        }
        __shared__ float red[32];
        red[lane] = acc;
        __syncthreads();
        if (lane == 0) {
            float s = 0.0f;
            #pragma unroll
            for (int i = 0; i < 32; ++i) s += red[i];
            ws[WS_DX2] = s;
        }
    }
}

// ---------------------------------------------------------------------------
// Kernel 2: per-(s,p) SPD solve. 8192 blocks x 64 threads.
// XtX (16 KB) is pulled into LDS by the Tensor Data Mover with hardware
// padding: pad_interval = 64 DWORDs, pad_amount = 1 DWORD => a 64x65 layout
// in LDS, so the gram-build reads hit bank (row+col)%64 (conflict-free),
// with zero VGPR traffic for the staging. TENSORcnt tracks completion.
// Then: gram = diag(m) XtX diag(m) + beta*I (SPD, Schur product theorem),
// in-place Cholesky + fwd/back substitution, write w(s,p,:) to workspace.
// ---------------------------------------------------------------------------
__global__ void k_solve(const float* __restrict__ G,
                        float* __restrict__ ws) {
    const int s   = blockIdx.x >> 6;
    const int p   = blockIdx.x & 63;
    const int tid = threadIdx.x;             // 0..63 (2 waves)
    const float* XtX  = ws + WS_XTX;
    const float* Xtdx = ws + WS_XTDX;
    float*       Wout = ws + WS_W;

    __shared__ float XtXl[D_DIM][D_DIM + 1]; // filled by TDM (padded)
    __shared__ float A[D_DIM][D_DIM + 1];
    __shared__ float mv[D_DIM];
    __shared__ float yv[D_DIM];

#if defined(HAVE_TDM)
    if (tid < 32) {                          // one DMA, issued by wave 0
        const unsigned long long ga = (unsigned long long)(uintptr_t)XtX;
        const unsigned int lds = (unsigned int)(uintptr_t)&XtXl[0][0];
        v4u g0;
        g0.x = 1u;                                   // count=1 valid descriptor
        g0.y = lds;                                  // lds_addr (bytes)
        g0.z = (unsigned int)ga;                     // global_addr[31:0]
        g0.w = ((unsigned int)(ga >> 32) & 0x1FFFFFFu) | (2u << 30); // [56:32] | type=2
        v8i g1;
        g1[0] = (2 << 16)      // data_size = 4 bytes
              | (1 << 20)      // pad_enable (load only)
              | (5 << 22);     // pad_interval: every 64 DWORDs; pad_amount=0 -> 1 DWORD
        g1[1] = (int)(64u << 16);   // tensor_dim0 = 64 (bits 79:48, low half)
        g1[2] = (int)(64u << 16);   // tensor_dim1 = 64 (bits 111:80, low half)
        g1[3] = (int)(64u << 16);   // tile_dim0  = 64 (bits 127:112)
        g1[4] = 64;                 // tile_dim1  = 64 (bits 143:128); tile_dim2=0
        g1[5] = 64;                 // tensor_dim0_stride = 64 (bits 207:160, low)
        g1[6] = 0;
        g1[7] = 0;
        v4i z4 = {0, 0, 0, 0};
#  if __clang_major__ >= 23
        v8i z8 = {0, 0, 0, 0, 0, 0, 0, 0};
        __builtin_amdgcn_tensor_load_to_lds(g0, g1, z4, z4, z8, 0);
#  else
        __builtin_amdgcn_tensor_load_to_lds(g0, g1, z4, z4, 0);
#  endif
        __builtin_amdgcn_s_wait_tensorcnt(0);
    }
#else
    for (int j = tid; j < D_DIM * D_DIM; j += 64)
        XtXl[j >> 6][j & 63] = XtX[j];
#endif

    const size_t gbase = (size_t)s * (D_DIM * D_DIM);
    __builtin_prefetch(&G[gbase + (size_t)tid * D_DIM + p], 0, 1);
    mv[tid] = G[gbase + (size_t)tid * D_DIM + p];   // mask m[i] = G[s,i,p]
    __syncthreads();

    const float mi = mv[tid];
    for (int j = 0; j < D_DIM; ++j)
        A[tid][j] = mi * mv[j] * XtXl[tid][j];
    A[tid][tid] += BETA;
    yv[tid] = mi * Xtdx[tid * D_DIM + p];
    __syncthreads();

    // In-place Cholesky (lower triangle), column-k broadcast pattern.
    for (int k = 0; k < D_DIM; ++k) {
        if (tid == k) A[k][k] = sqrtf(A[k][k]);
        __syncthreads();
        if (tid > k) A[tid][k] = A[tid][k] / A[k][k];
        __syncthreads();
        if (tid > k) {
            const float lik = A[tid][k];
            for (int j = k + 1; j <= tid; ++j)
                A[tid][j] -= lik * A[j][k];
        }
        __syncthreads();
    }
    // Forward substitution: L y = rhs
    for (int k = 0; k < D_DIM; ++k) {
        if (tid == k) yv[k] /= A[k][k];
        __syncthreads();
        if (tid > k) yv[tid] -= A[tid][k] * yv[k];
        __syncthreads();
    }
    // Backward substitution: L^T w = y
    for (int k = D_DIM - 1; k >= 0; --k) {
        if (tid == k) yv[k] /= A[k][k];
        __syncthreads();
        if (tid < k) yv[tid] -= A[k][tid] * yv[k];
        __syncthreads();
    }
    Wout[gbase + p * D_DIM + tid] = yv[tid];
}

// ---------------------------------------------------------------------------
// Kernel 3: per-sample energy. 128 blocks x 128 threads (4 waves).
// W[s] (16 KB) is staged into padded LDS via global_load_async_to_lds_b128
// (per-lane 16B direct-to-LDS copies, ASYNCcnt-tracked, no VGPR bounce).
// e1 = <XtX, W W^T>_F via WMMA f32 16x16x4 fed from LDS;
// e2 = <W, Xtdx>_F; prior = sum(G[s]).
// out[s] = (e1 - 2 e2 + ||dx||^2)/(N*D) + 0.1 * prior/D.
// ---------------------------------------------------------------------------
__global__ void k_energy(const float* __restrict__ G,
                         float* __restrict__ ws,
                         float* __restrict__ out) {
    const int s   = blockIdx.x;
    const int tid = threadIdx.x;             // 0..127
    const float* XtX  = ws + WS_XTX;
    const float* Xtdx = ws + WS_XTDX;
    const float* Wsrc = ws + WS_W + (size_t)s * (D_DIM * D_DIM);

    __shared__ float Wl[D_DIM][D_DIM + 1];   // padded: row stride 260 bytes
    __shared__ float r_e1, r_e2, r_pr;
    if (tid == 0) { r_e1 = 0.0f; r_e2 = 0.0f; r_pr = 0.0f; }

    // Async stage: 1024 x b128 chunks, 8 per lane. Chunk c covers row c>>4,
    // columns (c&15)*4 .. +3. LDS byte addr = base + row*260 + (c&15)*16.
    {
        const unsigned int lds_base = (unsigned int)(uintptr_t)&Wl[0][0];
        #pragma unroll
        for (int i = 0; i < 8; ++i) {
            const int c = tid + i * 128;
            const unsigned int ldsaddr =
                lds_base + (unsigned int)((c >> 4) * 260 + (c & 15) * 16);
            const unsigned int voff = (unsigned int)c * 16u;
            asm volatile("global_load_async_to_lds_b128 %0, %1, %2"
                         :: "v"(ldsaddr), "v"(voff), "s"(Wsrc)
                         : "memory");
        }
#if defined(__has_builtin) && __has_builtin(__builtin_amdgcn_s_wait_asynccnt)
        __builtin_amdgcn_s_wait_asynccnt(0);
#else
        asm volatile("s_wait_asynccnt 0x0" ::: "memory");
#endif
    }
    __syncthreads();

    const int wave = tid >> 5;               // 0..3 = row-block ti
    const int lane = tid & 31;
    const int half = lane >> 4;
    const int lm   = lane & 15;
    const int ti   = wave;

    float e1 = 0.0f;
    for (int tj = 0; tj < 4; ++tj) {
        v8f c = {};
        #pragma unroll
        for (int kb = 0; kb < D_DIM; kb += 4) {
            const int k0 = kb + 2 * half;
            v2f a, b;
            a.x = Wl[ti * 16 + lm][k0];
            a.y = Wl[ti * 16 + lm][k0 + 1];
            b.x = Wl[tj * 16 + lm][k0];      // (W^T)[k][n] = W[n][k]
            b.y = Wl[tj * 16 + lm][k0 + 1];
            c = __builtin_amdgcn_wmma_f32_16x16x4_f32(
                    false, a, false, b, (short)0, c, false, false);
        }
        #pragma unroll
        for (int r = 0; r < 8; ++r) {
            const int m = ti * 16 + r + 8 * half;
            const int n = tj * 16 + lm;
            e1 += c[r] * XtX[m * D_DIM + n];
        }
    }

    float e2 = 0.0f, pr = 0.0f;
    const size_t gbase = (size_t)s * (D_DIM * D_DIM);
    for (int idx = tid; idx < D_DIM * D_DIM; idx += 128) {
        e2 += Wl[idx >> 6][idx & 63] * Xtdx[idx];
        pr += G[gbase + idx];
    }

    atomicAdd(&r_e1, e1);    // ds_add_f32
    atomicAdd(&r_e2, e2);
    atomicAdd(&r_pr, pr);
    __syncthreads();

    if (tid == 0) {
        const float dx2 = ws[WS_DX2];
        const float mse = (r_e1 - 2.0f * r_e2 + dx2) /
                          ((float)N_ROWS * (float)D_DIM);
        out[s] = mse + PRIOR_LAMBDA * (r_pr / (float)D_DIM);
    }
}

// ---------------------------------------------------------------------------
extern "C" void kernel_launch(void* const* d_in, const int* in_sizes, int n_in,
                              void* d_out, int out_size, void* d_ws, size_t ws_size,
                              hipStream_t stream) {
    const float* G  = (const float*)d_in[0];   // [128,64,64]
    const float* x  = (const float*)d_in[1];   // [2048,64]
    const float* dx = (const float*)d_in[2];   // [2048,64]
    float* out = (float*)d_out;                // [128]
    float* ws  = (float*)d_ws;                 // needs ~2.15 MB

    // 1) XtX / Xtdx / ||dx||^2
    k_precompute<<<33, 32, 0, stream>>>(x, dx, ws);
    // 2) 8192 independent 64x64 SPD solves (TDM-staged XtX in LDS)
    k_solve<<<S_DIM * D_DIM, D_DIM, 0, stream>>>(G, ws);
    // 3) per-sample energy via trace identity + WMMA (async-staged W in LDS)
    k_energy<<<S_DIM, 128, 0, stream>>>(G, ws, out);
}